// GNNFiLMLayer_78091095376253
// MI455X (gfx1250) — compile-verified
//
#include <hip/hip_runtime.h>
#include <hip/hip_bf16.h>

// ---------------------------------------------------------------------------
// GNN-FiLM layer for MI455X (gfx1250, wave32, WMMA fp32 path).
//
// Phase 0 (pack):      re-layout [W_rel|film_rel] and loop_w into WMMA
//                      B-fragment order -> inner GEMM loop becomes
//                      global_load_b64 (immediate offset) + v_wmma only.
// Phase 1 (WMMA fp32): hWF[r][n][0:384] = feat @ [W_rel[r] | film_rel[r]]
// Phase 2 (WMMA fp32): out = feat @ loop_w + h_bias
// Phase 3 (edge):      out[dst] += relu(gamma(dst)*m(src) + beta(dst))
// ---------------------------------------------------------------------------

typedef float fx2 __attribute__((ext_vector_type(2)));
typedef float fx4 __attribute__((ext_vector_type(4)));
typedef float fx8 __attribute__((ext_vector_type(8)));

#define IN_FEAT    128
#define OUT_FEAT   128
#define NUM_RELS   4
#define N_NODES    50000
#define N_EDGES    800000
#define M_TILES    (N_NODES / 16)          // 3125, exact
#define KSTEPS     (IN_FEAT / 4)           // 32 wmma_f32_16x16x4 per 16x16 tile
#define WPB        8                       // waves per block (256 threads)
#define COMB       (OUT_FEAT * 3)          // 384: [m | gamma | beta]
#define NT_REL     (COMB / 16)             // 24 N-tiles per relation
#define NT_LOOP    (OUT_FEAT / 16)         // 8 N-tiles for loop_w

// Packed-B geometry: one tile = KSTEPS steps x 32 lanes x float2 = 2048 floats
#define TILE_F     (KSTEPS * 32 * 2)                 // 2048 floats / tile
#define REL_PACK_F (NT_REL * TILE_F)                 // 49152 floats / relation
#define PACK_REL_TOTAL_F (NUM_RELS * REL_PACK_F)     // 196608 floats
#define PACK_LOOP_OFF_F  PACK_REL_TOTAL_F
#define PACK_TOTAL_F     (PACK_REL_TOTAL_F + NT_LOOP * TILE_F)   // 212992 floats
#define HWF_OFF_F        262144                      // 1 MB in: hWF region start
#define PACK_PAIRS       (PACK_TOTAL_F / 2)          // 106496 float2 elements

// ---------------------------------------------------------------------------
// Phase 0: pack B matrices into fragment order.
// packed[tile][s][lane] = { B[4s+2kh][col0+n], B[4s+2kh+1][col0+n] }
//   with lane = n + 16*kh  (fp32 WMMA B 4x16 layout, ISA 7.12.2)
// ---------------------------------------------------------------------------
__global__ __launch_bounds__(256)
void pack_b_kernel(const float* __restrict__ W_rel,    // [R][128][128]
                   const float* __restrict__ film_rel, // [R][128][256]
                   const float* __restrict__ loop_w,   // [128][128]
                   float* __restrict__ packed)
{
    const int idx = blockIdx.x * 256 + threadIdx.x;    // one float2 per thread
    if (idx >= PACK_PAIRS) return;

    const int p    = idx & 1023;       // s*32 + lane within tile
    const int s    = p >> 5;
    const int lane = p & 31;
    const int n    = lane & 15;
    const int kh   = lane >> 4;
    const int k    = 4 * s + 2 * kh;

    const float* srcp;
    int ldb;
    if (idx < PACK_REL_TOTAL_F / 2) {
        const int t    = idx >> 10;    // r*NT_REL + nt
        const int r    = t / NT_REL;
        const int nt   = t - r * NT_REL;
        const int col0 = nt * 16;
        if (col0 < OUT_FEAT) {
            srcp = W_rel + r * (IN_FEAT * OUT_FEAT) + k * OUT_FEAT + col0 + n;
            ldb  = OUT_FEAT;
        } else {
            srcp = film_rel + r * (IN_FEAT * 2 * OUT_FEAT) + k * (2 * OUT_FEAT)
                 + (col0 - OUT_FEAT) + n;
            ldb  = 2 * OUT_FEAT;
        }
    } else {
        const int nt = (idx - PACK_REL_TOTAL_F / 2) >> 10;
        srcp = loop_w + k * OUT_FEAT + nt * 16 + n;
        ldb  = OUT_FEAT;
    }
    fx2 v;
    v.x = srcp[0];
    v.y = srcp[ldb];
    *(fx2*)(packed + 2 * idx) = v;
}

// ---------------------------------------------------------------------------
// Phase 1: per-relation fused projection. One wave = one 16-row M tile,
// A slab resident in 64 VGPRs, 24 N-tiles swept from packed B.
// ---------------------------------------------------------------------------
__global__ __launch_bounds__(256)
void film_proj_kernel(const float* __restrict__ A,        // feat [N_NODES][128]
                      const float* __restrict__ packedB,  // fragment-ordered B
                      float* __restrict__ Out)            // hWF [R][N_NODES][384]
{
    const int r     = blockIdx.y;
    const int wave  = threadIdx.x >> 5;
    const int lane  = threadIdx.x & 31;
    const int mtile = blockIdx.x * WPB + wave;
    if (mtile >= M_TILES) return;            // wave-uniform: EXEC stays full

    const int mrow = lane & 15;
    const int kh   = lane >> 4;
    const int n    = lane & 15;

    fx2 a[KSTEPS];
    const float* arow = A + (size_t)(mtile * 16 + mrow) * IN_FEAT + 2 * kh;
#pragma unroll
    for (int s = 0; s < KSTEPS; ++s)
        a[s] = *(const fx2*)(arow + 4 * s);

    const float* prel  = packedB + (size_t)r * REL_PACK_F + lane * 2;
    float* outbase     = Out + ((size_t)r * N_NODES + (size_t)mtile * 16) * COMB;

    for (int nt = 0; nt < NT_REL; ++nt) {
        const float* bb = prel + nt * TILE_F;   // single base; s*256B immediates
        fx8 c = {};
#pragma unroll
        for (int s = 0; s < KSTEPS; ++s) {
            const fx2 b = *(const fx2*)(bb + s * 64);
            c = __builtin_amdgcn_wmma_f32_16x16x4_f32(
                    false, a[s], false, b, (short)0, c, false, false);
        }
        float* op = outbase + nt * 16 + n;
#pragma unroll
        for (int i = 0; i < 8; ++i)
            op[(i + 8 * kh) * COMB] = c[i];
    }
}

// ---------------------------------------------------------------------------
// Phase 2: out = feat @ loop_w + h_bias   (initializes the segment sum)
// ---------------------------------------------------------------------------
__global__ __launch_bounds__(256)
void self_loop_kernel(const float* __restrict__ A,        // feat [N_NODES][128]
                      const float* __restrict__ packedB,  // fragment-ordered B
                      const float* __restrict__ bias,     // h_bias [128]
                      float* __restrict__ Out)            // out [N_NODES][128]
{
    const int wave  = threadIdx.x >> 5;
    const int lane  = threadIdx.x & 31;
    const int mtile = blockIdx.x * WPB + wave;
    if (mtile >= M_TILES) return;

    const int mrow = lane & 15;
    const int kh   = lane >> 4;
    const int n    = lane & 15;

    fx2 a[KSTEPS];
    const float* arow = A + (size_t)(mtile * 16 + mrow) * IN_FEAT + 2 * kh;
#pragma unroll
    for (int s = 0; s < KSTEPS; ++s)
        a[s] = *(const fx2*)(arow + 4 * s);

    const float* pl = packedB + PACK_LOOP_OFF_F + lane * 2;
    float* outbase  = Out + (size_t)mtile * 16 * OUT_FEAT;

    for (int nt = 0; nt < NT_LOOP; ++nt) {
        const float* bb = pl + nt * TILE_F;
        fx8 c = {};
#pragma unroll
        for (int s = 0; s < KSTEPS; ++s) {
            const fx2 b = *(const fx2*)(bb + s * 64);
            c = __builtin_amdgcn_wmma_f32_16x16x4_f32(
                    false, a[s], false, b, (short)0, c, false, false);
        }
        const float bv = bias[nt * 16 + n];
        float* op = outbase + nt * 16 + n;
#pragma unroll
        for (int i = 0; i < 8; ++i)
            op[(i + 8 * kh) * OUT_FEAT] = c[i] + bv;
    }
}

// ---------------------------------------------------------------------------
// Phase 3: edge FiLM + scatter-add. One wave per edge, float4 per lane
// (32 lanes x 4 = 128 features, 512B contiguous gathers).
// ---------------------------------------------------------------------------
__global__ __launch_bounds__(256)
void edge_film_kernel(const float* __restrict__ hWF,   // [R][N_NODES][384]
                      const int* __restrict__ src,
                      const int* __restrict__ dst,
                      const int* __restrict__ etypes,
                      float* __restrict__ out)          // [N_NODES][128]
{
    const int wave = threadIdx.x >> 5;
    const int lane = threadIdx.x & 31;
    const int e    = blockIdx.x * WPB + wave;
    if (e >= N_EDGES) return;

    const int s = src[e];
    const int d = dst[e];
    const int r = etypes[e];

    const size_t bs = ((size_t)r * N_NODES + (size_t)s) * COMB;
    const size_t bd = ((size_t)r * N_NODES + (size_t)d) * COMB;
    const int    f  = lane * 4;

    const fx4 m = *(const fx4*)(hWF + bs + f);
    const fx4 g = *(const fx4*)(hWF + bd + OUT_FEAT + f);
    const fx4 b = *(const fx4*)(hWF + bd + 2 * OUT_FEAT + f);

    fx4 v;
    v.x = fmaxf(g.x * m.x + b.x, 0.0f);
    v.y = fmaxf(g.y * m.y + b.y, 0.0f);
    v.z = fmaxf(g.z * m.z + b.z, 0.0f);
    v.w = fmaxf(g.w * m.w + b.w, 0.0f);

    float* o = out + (size_t)d * OUT_FEAT + f;
    unsafeAtomicAdd(o + 0, v.x);   // global_atomic_add_f32 (no CAS loop)
    unsafeAtomicAdd(o + 1, v.y);
    unsafeAtomicAdd(o + 2, v.z);
    unsafeAtomicAdd(o + 3, v.w);
}

// ---------------------------------------------------------------------------
extern "C" void kernel_launch(void* const* d_in, const int* in_sizes, int n_in,
                              void* d_out, int out_size, void* d_ws, size_t ws_size,
                              hipStream_t stream) {
    const float* feat     = (const float*)d_in[0];
    const int*   src      = (const int*)  d_in[1];
    const int*   dst      = (const int*)  d_in[2];
    const int*   etypes   = (const int*)  d_in[3];
    const float* W_rel    = (const float*)d_in[4];
    const float* film_rel = (const float*)d_in[5];
    const float* h_bias   = (const float*)d_in[6];
    const float* loop_w   = (const float*)d_in[7];
    float*       out      = (float*)d_out;

    float* packedB = (float*)d_ws;                 // 832 KB fragment-ordered B
    float* hWF     = (float*)d_ws + HWF_OFF_F;     // 4*50000*384 floats = 307.2 MB

    // Phase 0: pack B (tiny; L2-resident afterwards).
    const int pack_blocks = (PACK_PAIRS + 255) / 256;            // 416
    pack_b_kernel<<<dim3(pack_blocks), 256, 0, stream>>>(
        W_rel, film_rel, loop_w, packedB);

    const int gemm_blocks = (M_TILES + WPB - 1) / WPB;           // 391

    // Phase 1: projections for all 4 relations into workspace.
    film_proj_kernel<<<dim3(gemm_blocks, NUM_RELS), 256, 0, stream>>>(
        feat, packedB, hWF);

    // Phase 2: initialize out with self-loop + bias.
    self_loop_kernel<<<dim3(gemm_blocks, 1), 256, 0, stream>>>(
        feat, packedB, h_bias, out);

    // Phase 3: edge FiLM + atomic segment sum.
    const int edge_blocks = (N_EDGES + WPB - 1) / WPB;           // 100000
    edge_film_kernel<<<dim3(edge_blocks, 1), 256, 0, stream>>>(
        hWF, src, dst, etypes, out);
}